// BosonSamplerTorch_26611617366280
// MI455X (gfx1250) — compile-verified
//
#include <hip/hip_runtime.h>
#include <math.h>

typedef float v2f __attribute__((ext_vector_type(2)));
typedef float v8f __attribute__((ext_vector_type(8)));

#define MODES      64
#define NPH        12
#define NSUB       4096          // 2^NPH
#define MU_C       0.9f
#define ALPHA_C    0.1f
#define BETA_C     0.5f
#define DARK_C     1e-5f

// One block per batch element. 128 threads = 4 waves (wave32).
// Each wave handles 1024 subsets = 64 tiles of 16 subsets.
// WMMA computes D[i, s] = sum_j A[i][j] * mask[s][j] with the complex matrix
// (re/im) in the loop-invariant A-operand and the subset masks in B.
// C-layout then gives each lane its own subset's row-sums in registers:
//   lane L -> subset tb + (L&15); VGPR r -> row i=r (lanes 0-15) / i=r+8 (16-31)
// so the Ryser per-subset product over i is in-lane, with a single xor-16
// exchange to merge the two half-products.
__global__ __launch_bounds__(128)
void boson_sampler_kernel(const float* __restrict__ U_re,
                          const float* __restrict__ U_im,
                          const int*   __restrict__ input_modes,
                          const int*   __restrict__ output_modes,
                          float*       __restrict__ out)
{
    const int b    = blockIdx.x;
    const int tid  = threadIdx.x;
    const int lane = tid & 31;
    const int wv   = tid >> 5;        // wave id 0..3
    const int li   = lane & 15;       // lane within 16-half
    const int half = lane >> 4;       // 0: lanes 0-15, 1: lanes 16-31

    // A stored transposed: sAreT[j*16 + i] = Re A[i][j]
    __shared__ float sAreT[NPH * 16];
    __shared__ float sAimT[NPH * 16];
    __shared__ float sWred[12];       // [0..3] classical, [4..7] perm re, [8..11] perm im
    __shared__ float sNl;

    const int* inm  = input_modes  + b * NPH;
    const int* outm = output_modes + b * NPH;

    // ---- Phase 1: gather A into LDS + classical probability partial product
    float lp = 1.0f;
    for (int e = tid; e < NPH * NPH; e += blockDim.x) {
        int i = e / NPH;
        int j = e - i * NPH;
        int r = inm[i];
        int c = outm[j];
        float ar = U_re[r * MODES + c];
        float ai = U_im[r * MODES + c];
        sAreT[j * 16 + i] = ar;
        sAimT[j * 16 + i] = ai;
        lp *= (ar * ar + ai * ai);
    }
    #pragma unroll
    for (int m = 16; m >= 1; m >>= 1) lp *= __shfl_xor(lp, m, 32);
    if (lane == 0) sWred[wv] = lp;

    // ---- nonlinear factor: prod over distinct output modes of (1+alpha*cnt)^-beta
    if (tid == 0) {
        float nl = 1.0f;
        for (int j = 0; j < NPH; ++j) {
            int mj = outm[j];
            int cnt = 0;
            bool first = true;
            for (int k = 0; k < NPH; ++k) {
                int mk = outm[k];
                cnt += (mk == mj) ? 1 : 0;
                if (k < j && mk == mj) first = false;
            }
            if (first) nl *= powf(1.0f + ALPHA_C * (float)cnt, -BETA_C);
        }
        sNl = nl;
    }
    __syncthreads();

    // ---- Phase 2: loop-invariant A-operands (the complex matrix), per k-step.
    // A layout (16x4 f32): lane L -> row M = L&15, VGPR r -> K = r + 2*(L>>4).
    // A_op[m=i][k=j] = A[i][j]; zero-pad rows i >= 12.
    v2f are[3], aim[3];
    #pragma unroll
    for (int ks = 0; ks < 3; ++ks) {
        #pragma unroll
        for (int r = 0; r < 2; ++r) {
            int j = ks * 4 + 2 * half + r;
            float vr = 0.0f, vi = 0.0f;
            if (li < NPH) {
                vr = sAreT[j * 16 + li];
                vi = sAimT[j * 16 + li];
            }
            are[ks][r] = vr;
            aim[ks][r] = vi;
        }
    }

    // ---- Phase 3: Ryser sum over this wave's 1024 subsets (64 tiles of 16)
    float accRe = 0.0f, accIm = 0.0f;
    const int wbase = wv * (NSUB / 4);

    for (int t = 0; t < 64; ++t) {
        const int tb   = wbase + t * 16;
        const int scol = tb + li;     // subset carried by this lane's B-column
        v8f cre = {};
        v8f cim = {};
        #pragma unroll
        for (int ks = 0; ks < 3; ++ks) {
            // B operand (4x16 f32 masks): lane L -> N = L&15, VGPR r -> K = r+2*(L>>4)
            const int j0 = ks * 4 + 2 * half;
            v2f bm;
            bm[0] = (float)((scol >> j0) & 1);
            bm[1] = (float)((scol >> (j0 + 1)) & 1);
            cre = __builtin_amdgcn_wmma_f32_16x16x4_f32(
                    false, are[ks], false, bm, (short)0, cre, false, false);
            cim = __builtin_amdgcn_wmma_f32_16x16x4_f32(
                    false, aim[ks], false, bm, (short)0, cim, false, false);
        }
        // In-lane complex product over this lane's rows:
        //   lanes 0-15: i = 0..7 (VGPRs 0-7); lanes 16-31: i = 8..11 (VGPRs 0-3,
        //   VGPRs 4-7 are padded rows 12-15 -> replaced by complex identity).
        float pr = cre[0];
        float pi = cim[0];
        #pragma unroll
        for (int r = 1; r < 8; ++r) {
            float vr = cre[r];
            float vi = cim[r];
            if (r >= 4 && half) { vr = 1.0f; vi = 0.0f; }  // cndmask, no branch
            float nr = pr * vr - pi * vi;
            float ni = pr * vi + pi * vr;
            pr = nr; pi = ni;
        }
        // Merge the two half-products (rows 0-7 with rows 8-11): single xor-16.
        float qr = __shfl_xor(pr, 16, 32);
        float qi = __shfl_xor(pi, 16, 32);
        float fr = pr * qr - pi * qi;
        float fi = pr * qi + pi * qr;
        // Accumulate: 16 subsets per tile, one per lower-half lane.
        float sgn = (__popc(scol) & 1) ? -1.0f : 1.0f;     // n=12 even: + iff |S| even
        sgn = half ? 0.0f : sgn;                            // upper half contributes 0
        accRe += sgn * fr;
        accIm += sgn * fi;
    }

    // ---- Phase 4: reduce permanent across lanes and waves, finalize
    #pragma unroll
    for (int m = 16; m >= 1; m >>= 1) {
        accRe += __shfl_xor(accRe, m, 32);
        accIm += __shfl_xor(accIm, m, 32);
    }
    if (lane == 0) {
        sWred[4 + wv] = accRe;
        sWred[8 + wv] = accIm;
    }
    __syncthreads();

    if (tid == 0) {
        float cl = sWred[0] * sWred[1] * sWred[2] * sWred[3];
        float pr = sWred[4] + sWred[5] + sWred[6] + sWred[7];
        float pi = sWred[8] + sWred[9] + sWred[10] + sWred[11];
        float nl = sNl;
        float prob = MU_C * (nl * nl) * (pr * pr + pi * pi)
                   + (1.0f - MU_C) * cl
                   + DARK_C * (float)MODES;
        out[b] = prob;
    }
}

extern "C" void kernel_launch(void* const* d_in, const int* in_sizes, int n_in,
                              void* d_out, int out_size, void* d_ws, size_t ws_size,
                              hipStream_t stream) {
    (void)in_sizes; (void)n_in; (void)d_ws; (void)ws_size;
    const float* U_re = (const float*)d_in[0];
    const float* U_im = (const float*)d_in[1];
    const int*   inm  = (const int*)d_in[2];
    const int*   outm = (const int*)d_in[3];
    float* out = (float*)d_out;

    dim3 grid((unsigned)out_size);   // 1024 batches, one block each
    dim3 block(128);                 // 4 wave32
    hipLaunchKernelGGL(boson_sampler_kernel, grid, block, 0, stream,
                       U_re, U_im, inm, outm, out);
}